// GraphSAGE_69475390980336
// MI455X (gfx1250) — compile-verified
//
#include <hip/hip_runtime.h>
#include <math.h>

typedef float v2f __attribute__((ext_vector_type(2)));
typedef float v8f __attribute__((ext_vector_type(8)));

namespace {
constexpr int kN  = 200000;
constexpr int kF  = 602;
constexpr int kF2 = kF / 2;   // 301 float2 per feature row (rows are 8B-aligned: 602*4 % 8 == 0)
constexpr int kD  = 128;
constexpr int kH  = 256;
constexpr int kC  = 41;
constexpr int kB  = 1024;
constexpr int kS1 = 25;
constexpr int kS2 = 10;
constexpr int kH2 = 2 * kH;   // 512
}

// ---------------- sampling ----------------
__global__ void sample_hop0_k(const int* __restrict__ x, const int* __restrict__ adj,
                              const int* __restrict__ perm0, int* __restrict__ hop0) {
    int i = blockIdx.x * blockDim.x + threadIdx.x;
    if (i >= kB * kS2) return;
    int b = i / kS2, j = i % kS2;
    hop0[i] = adj[(long long)x[b] * kD + perm0[j]];
}

__global__ void sample_hop1_k(const int* __restrict__ hop0, const int* __restrict__ adj,
                              const int* __restrict__ perm1, int* __restrict__ hop1) {
    int i = blockIdx.x * blockDim.x + threadIdx.x;
    if (i >= kB * kS2 * kS1) return;
    int p = i / kS1, j = i % kS1;
    hop1[i] = adj[(long long)hop0[p] * kD + perm1[j]];
}

// ---------------- gathers (float2-vectorized; feature rows are 8B-aligned) ----------------
// out[r, :] = feat[ids[r], :]
__global__ void gather_rows_k(const float* __restrict__ feat, const int* __restrict__ ids,
                              float* __restrict__ out) {
    const int r = blockIdx.x;
    const v2f* __restrict__ src = (const v2f*)(feat + (long long)ids[r] * kF);
    v2f* __restrict__ dst = (v2f*)(out + (long long)r * kF);
    for (int f = threadIdx.x; f < kF2; f += blockDim.x)
        dst[f] = src[f];
}

// out[g, :] = mean_j feat[ids[g*S + j], :]   (S <= 32)
__global__ void gather_mean_k(const float* __restrict__ feat, const int* __restrict__ ids,
                              float* __restrict__ out, int S, float inv) {
    const int g = blockIdx.x;
    __shared__ int nid[32];
    if (threadIdx.x < (unsigned)S) nid[threadIdx.x] = ids[g * S + threadIdx.x];
    __syncthreads();
    // prefetch the S neighbor rows (19 x 128B cachelines cover 2408 B each)
    for (int t = threadIdx.x; t < S * 19; t += blockDim.x) {
        int j = t / 19, c = t % 19;
        __builtin_prefetch(feat + (long long)nid[j] * kF + c * 32, 0, 3);
    }
    v2f* __restrict__ dst = (v2f*)(out + (long long)g * kF);
    for (int f = threadIdx.x; f < kF2; f += blockDim.x) {
        v2f s; s.x = 0.0f; s.y = 0.0f;
        #pragma unroll 5
        for (int j = 0; j < S; ++j) {
            v2f v = *(const v2f*)(feat + (long long)nid[j] * kF + 2 * f);
            s.x += v.x; s.y += v.y;
        }
        s.x *= inv; s.y *= inv;
        dst[f] = s;
    }
}

// out[g, f] = mean_j in[(g*S + j), f]  (contiguous input, F even)
__global__ void mean_rows_k(const float* __restrict__ in, float* __restrict__ out,
                            int S, int F, float inv) {
    const int g = blockIdx.x;
    const int F2 = F >> 1;
    v2f* __restrict__ dst = (v2f*)(out + (long long)g * F);
    for (int f = threadIdx.x; f < F2; f += blockDim.x) {
        v2f s; s.x = 0.0f; s.y = 0.0f;
        for (int j = 0; j < S; ++j) {
            v2f v = *(const v2f*)(in + ((long long)g * S + j) * F + 2 * f);
            s.x += v.x; s.y += v.y;
        }
        s.x *= inv; s.y *= inv;
        dst[f] = s;
    }
}

// ---------------- WMMA GEMM: Y[:, coloff + 0..Nout) = relu?(X @ W^T + bias) ----------------
// X: [M, K] row-major, W: [Nout, K] row-major, Y: [M, ldy] row-major.
// M must be a multiple of 16. One wave (32 lanes) per 16x16 tile of C.
// grid = (M/16, ceil(Nout/16)), block = 32.
__global__ void __launch_bounds__(32)
gemm16_wmma_f32(const float* __restrict__ X, const float* __restrict__ W,
                const float* __restrict__ bias, float* __restrict__ Y,
                int K, int ldy, int coloff, int Nout, int do_relu) {
    const int lane = threadIdx.x & 31;
    const int half = lane >> 4;      // 0: K pair {0,1}, 1: K pair {2,3}
    const int l16  = lane & 15;
    const int m0 = blockIdx.x << 4;
    const int n0 = blockIdx.y << 4;

    const int arow  = m0 + l16;                    // A: 16x4, M striped across lanes 0-15 / 16-31
    const int bcol  = n0 + l16;                    // B: 4x16, N striped across lanes
    const int bclmp = (bcol < Nout) ? bcol : (Nout - 1);
    const float bmask = (bcol < Nout) ? 1.0f : 0.0f;

    const float* xp = X + (long long)arow * K + (half << 1);
    const float* wp = W + (long long)bclmp * K + (half << 1);

    v8f acc = {};
    int k = 0;

    if (n0 + 16 <= Nout) {
        // full tile: no column masking (wave-uniform branch; EXEC stays all ones)
        for (; k + 4 <= K; k += 4) {
            v2f a = *(const v2f*)(xp + k);
            v2f b = *(const v2f*)(wp + k);
            acc = __builtin_amdgcn_wmma_f32_16x16x4_f32(false, a, false, b,
                                                        (short)0, acc, false, false);
        }
    } else {
        for (; k + 4 <= K; k += 4) {
            v2f a = *(const v2f*)(xp + k);
            v2f b = *(const v2f*)(wp + k);
            b.x *= bmask;
            b.y *= bmask;
            acc = __builtin_amdgcn_wmma_f32_16x16x4_f32(false, a, false, b,
                                                        (short)0, acc, false, false);
        }
    }
    if (k < K) {   // tail (K % 4 != 0): zero-pad out-of-range K
        const int kk = k + (half << 1);
        v2f a; a.x = 0.0f; a.y = 0.0f;
        v2f b; b.x = 0.0f; b.y = 0.0f;
        if (kk < K) {
            a.x = X[(long long)arow * K + kk];
            b.x = bmask * W[(long long)bclmp * K + kk];
            if (kk + 1 < K) {
                a.y = X[(long long)arow * K + kk + 1];
                b.y = bmask * W[(long long)bclmp * K + kk + 1];
            }
        }
        acc = __builtin_amdgcn_wmma_f32_16x16x4_f32(false, a, false, b,
                                                    (short)0, acc, false, false);
    }

    // C/D layout: VGPR g -> row m0 + g + 8*half, col n0 + l16
    if (bcol < Nout) {
        const float bv = bias[bcol];
        #pragma unroll
        for (int g = 0; g < 8; ++g) {
            const int row = m0 + (half << 3) + g;
            float v = acc[g] + bv;
            if (do_relu) v = fmaxf(v, 0.0f);
            Y[(long long)row * ldy + coloff + bcol] = v;
        }
    }
}

// ---------------- layernorm over 512 ----------------
__global__ void __launch_bounds__(256)
layernorm_k(const float* __restrict__ in, const float* __restrict__ g,
            const float* __restrict__ b, float* __restrict__ out) {
    __shared__ float red[256];
    const int row = blockIdx.x;
    const int t = threadIdx.x;
    const float* p = in + (long long)row * kH2;
    float v0 = p[t], v1 = p[t + 256];
    red[t] = v0 + v1;
    __syncthreads();
    for (int s = 128; s > 0; s >>= 1) {
        if (t < s) red[t] += red[t + s];
        __syncthreads();
    }
    const float mu = red[0] * (1.0f / kH2);
    __syncthreads();
    const float d0 = v0 - mu, d1 = v1 - mu;
    red[t] = d0 * d0 + d1 * d1;
    __syncthreads();
    for (int s = 128; s > 0; s >>= 1) {
        if (t < s) red[t] += red[t + s];
        __syncthreads();
    }
    const float inv = 1.0f / sqrtf(red[0] * (1.0f / kH2) + 1e-12f);
    float* q = out + (long long)row * kH2;
    q[t]       = d0 * inv * g[t]       + b[t];
    q[t + 256] = d1 * inv * g[t + 256] + b[t + 256];
}

// ---------------- host ----------------
extern "C" void kernel_launch(void* const* d_in, const int* in_sizes, int n_in,
                              void* d_out, int out_size, void* d_ws, size_t ws_size,
                              hipStream_t stream) {
    (void)in_sizes; (void)n_in; (void)out_size; (void)ws_size;

    const int*   x         = (const int*)  d_in[0];
    const int*   adj       = (const int*)  d_in[1];
    const int*   perm0     = (const int*)  d_in[2];
    const int*   perm1     = (const int*)  d_in[3];
    const float* features  = (const float*)d_in[4];
    const float* w1a_self  = (const float*)d_in[5];
    const float* b1a_self  = (const float*)d_in[6];
    const float* w1a_neigh = (const float*)d_in[7];
    const float* b1a_neigh = (const float*)d_in[8];
    const float* w1b_self  = (const float*)d_in[9];
    const float* b1b_self  = (const float*)d_in[10];
    const float* w1b_neigh = (const float*)d_in[11];
    const float* b1b_neigh = (const float*)d_in[12];
    const float* w2_self   = (const float*)d_in[13];
    const float* b2_self   = (const float*)d_in[14];
    const float* w2_neigh  = (const float*)d_in[15];
    const float* b2_neigh  = (const float*)d_in[16];
    const float* ln_g      = (const float*)d_in[17];
    const float* ln_b      = (const float*)d_in[18];
    const float* w_cls     = (const float*)d_in[19];
    const float* b_cls     = (const float*)d_in[20];
    float* out = (float*)d_out;

    char* ws = (char*)d_ws;
    size_t off = 0;
    auto alloc = [&](size_t bytes) -> char* {
        char* p = ws + off;
        off += (bytes + 255) & ~(size_t)255;
        return p;
    };

    int*   hop0  = (int*)  alloc((size_t)kB * kS2 * sizeof(int));         // [10240]
    int*   hop1  = (int*)  alloc((size_t)kB * kS2 * kS1 * sizeof(int));   // [256000]
    float* Aseed = (float*)alloc((size_t)kB * kF * sizeof(float));        // features[x]
    float* Fh0   = (float*)alloc((size_t)kB * kS2 * kF * sizeof(float));  // features[hop0]
    float* M0    = (float*)alloc((size_t)kB * kF * sizeof(float));        // mean_j Fh0
    float* M1    = (float*)alloc((size_t)kB * kS2 * kF * sizeof(float));  // mean_k features[hop1]
    float* fm1   = (float*)alloc((size_t)kB * kS2 * kH2 * sizeof(float)); // featmap1 [10240,512]
    float* fm0   = (float*)alloc((size_t)kB * kH2 * sizeof(float));       // featmap0 [1024,512]
    float* x1    = (float*)alloc((size_t)kB * kH2 * sizeof(float));       // mean_j fm1
    float* cat   = (float*)alloc((size_t)kB * kH2 * sizeof(float));       // pre-layernorm
    float* lnO   = (float*)alloc((size_t)kB * kH2 * sizeof(float));       // post-layernorm

    // sampling
    sample_hop0_k<<<(kB * kS2 + 255) / 256, 256, 0, stream>>>(x, adj, perm0, hop0);
    sample_hop1_k<<<(kB * kS2 * kS1 + 255) / 256, 256, 0, stream>>>(hop0, adj, perm1, hop1);

    // gathers + means
    gather_rows_k<<<kB, 256, 0, stream>>>(features, x, Aseed);
    gather_rows_k<<<kB * kS2, 256, 0, stream>>>(features, hop0, Fh0);
    mean_rows_k  <<<kB, 256, 0, stream>>>(Fh0, M0, kS2, kF, 1.0f / kS2);
    gather_mean_k<<<kB * kS2, 256, 0, stream>>>(features, hop1, M1, kS1, 1.0f / kS1);

    const dim3 blk(32, 1, 1);
    // featmap1 = relu([Fh0 @ w1b_self^T + b | M1 @ w1b_neigh^T + b])
    gemm16_wmma_f32<<<dim3(kB * kS2 / 16, kH / 16), blk, 0, stream>>>(
        Fh0, w1b_self, b1b_self, fm1, kF, kH2, 0, kH, 1);
    gemm16_wmma_f32<<<dim3(kB * kS2 / 16, kH / 16), blk, 0, stream>>>(
        M1, w1b_neigh, b1b_neigh, fm1, kF, kH2, kH, kH, 1);
    // featmap0 = relu([Aseed @ w1a_self^T + b | M0 @ w1a_neigh^T + b])
    gemm16_wmma_f32<<<dim3(kB / 16, kH / 16), blk, 0, stream>>>(
        Aseed, w1a_self, b1a_self, fm0, kF, kH2, 0, kH, 1);
    gemm16_wmma_f32<<<dim3(kB / 16, kH / 16), blk, 0, stream>>>(
        M0, w1a_neigh, b1a_neigh, fm0, kF, kH2, kH, kH, 1);
    // x1 = mean over S2 of featmap1
    mean_rows_k<<<kB, 256, 0, stream>>>(fm1, x1, kS2, kH2, 1.0f / kS2);
    // cat = [fm0 @ w2_self^T + b | x1 @ w2_neigh^T + b]
    gemm16_wmma_f32<<<dim3(kB / 16, kH / 16), blk, 0, stream>>>(
        fm0, w2_self, b2_self, cat, kH2, kH2, 0, kH, 0);
    gemm16_wmma_f32<<<dim3(kB / 16, kH / 16), blk, 0, stream>>>(
        x1, w2_neigh, b2_neigh, cat, kH2, kH2, kH, kH, 0);
    // layernorm
    layernorm_k<<<kB, 256, 0, stream>>>(cat, ln_g, ln_b, lnO);
    // classifier: [1024,512] @ [41,512]^T + b  -> out [1024,41]
    gemm16_wmma_f32<<<dim3(kB / 16, (kC + 15) / 16), blk, 0, stream>>>(
        lnO, w_cls, b_cls, out, kH2, kC, 0, kC, 0);
}